// SimpleConvGCN_18700287606915
// MI455X (gfx1250) — compile-verified
//
#include <hip/hip_runtime.h>
#include <hip/hip_bf16.h>

typedef float v2f __attribute__((ext_vector_type(2)));
typedef float v8f __attribute__((ext_vector_type(8)));

// ---------------------------------------------------------------------------
// Node-feature GEMM via V_WMMA_F32_16X16X4_F32, K known at compile time so
// the K-loop fully unrolls.  One wave computes a 16 x (16*NT) output tile:
// the A fragment (2 VGPRs) is reused across NT WMMAs per K-step.
// Fused epilogue: C = A@B (needed for edge gathers) and H = C * dinv[row]^2
// (self-loop initialization of the aggregation buffer) — saves a full
// read-modify-write pass over the N*D activation tensor.
// A layout per ISA (16x4 f32): lane group g = lane>>4 holds K = {2g, 2g+1};
// B symmetric (col = lane&15); C/D: VGPR v holds row v + 8*g, col = lane&15.
// ---------------------------------------------------------------------------
template <int K, int NT>
__global__ void gemm_node_wmma(const float* __restrict__ A,
                               const float* __restrict__ B,
                               float* __restrict__ C,
                               float* __restrict__ H,
                               const float* __restrict__ dinv,
                               int M, int N) {
  int wave = (int)((blockIdx.x * blockDim.x + threadIdx.x) >> 5);
  int lane = (int)(threadIdx.x & 31);
  int colGroups = N / (16 * NT);
  int rowTiles = (M + 15) >> 4;
  if (wave >= rowTiles * colGroups) return;   // wave-uniform: EXEC all-1s

  int rt = wave / colGroups;
  int cg = wave - rt * colGroups;
  int g  = lane >> 4;
  int lr = lane & 15;

  int arow = rt * 16 + lr;  if (arow >= M) arow = M - 1;
  const float* Arow = A + (long)arow * K;
  int colbase = cg * 16 * NT;

  v8f acc[NT];
#pragma unroll
  for (int t = 0; t < NT; ++t) acc[t] = (v8f){};

#pragma unroll
  for (int k0 = 0; k0 < K; k0 += 4) {
    int ka = k0 + 2 * g;
    v2f a;
    a[0] = Arow[ka];
    a[1] = Arow[ka + 1];
#pragma unroll
    for (int t = 0; t < NT; ++t) {
      int bc = colbase + 16 * t + lr;
      v2f b;
      b[0] = B[(long)ka * N + bc];
      b[1] = B[(long)(ka + 1) * N + bc];
      acc[t] = __builtin_amdgcn_wmma_f32_16x16x4_f32(false, a, false, b,
                                                     (short)0, acc[t],
                                                     false, false);
    }
  }

  int crow0 = rt * 16 + 8 * g;
#pragma unroll
  for (int v = 0; v < 8; ++v) {
    int r = crow0 + v;
    if (r < M) {
      float w = dinv[r];
      float w2 = w * w;
#pragma unroll
      for (int t = 0; t < NT; ++t) {
        int cc = colbase + 16 * t + lr;
        float val = acc[t][v];
        C[(long)r * N + cc] = val;
        H[(long)r * N + cc] = val * w2;
      }
    }
  }
}

// ---------------------------------------------------------------------------
// Generic f32 WMMA GEMM (one 16x16 tile per wave) for the small head GEMM
// where N (1028) is not a multiple of 64.
// ---------------------------------------------------------------------------
__global__ void gemm_wmma_f32(const float* __restrict__ A,
                              const float* __restrict__ B,
                              float* __restrict__ C,
                              int M, int N, int K) {
  int wave = (int)((blockIdx.x * blockDim.x + threadIdx.x) >> 5);
  int lane = (int)(threadIdx.x & 31);
  int colTiles = (N + 15) >> 4;
  int rowTiles = (M + 15) >> 4;
  if (wave >= rowTiles * colTiles) return;

  int rt = wave / colTiles;
  int ct = wave - rt * colTiles;
  int g  = lane >> 4;
  int lr = lane & 15;

  int arow = rt * 16 + lr;  if (arow >= M) arow = M - 1;
  int bcol = ct * 16 + lr;  if (bcol >= N) bcol = N - 1;

  v8f acc = {};
  for (int k0 = 0; k0 < K; k0 += 4) {
    int ka = k0 + 2 * g;
    v2f a, b;
    a[0] = A[(long)arow * K + ka];
    a[1] = A[(long)arow * K + ka + 1];
    b[0] = B[(long)ka * N + bcol];
    b[1] = B[(long)(ka + 1) * N + bcol];
    acc = __builtin_amdgcn_wmma_f32_16x16x4_f32(false, a, false, b,
                                                (short)0, acc, false, false);
  }

  int ccol = ct * 16 + lr;
  if (ccol < N) {
    int crow0 = rt * 16 + 8 * g;
#pragma unroll
    for (int v = 0; v < 8; ++v) {
      int r = crow0 + v;
      if (r < M) C[(long)r * N + ccol] = acc[v];
    }
  }
}

// ---------------------------------------------------------------------------
// GCN support kernels
// ---------------------------------------------------------------------------
__global__ void init_deg(float* deg, int n) {
  int i = blockIdx.x * blockDim.x + threadIdx.x;
  if (i < n) deg[i] = 1.0f;                 // self-loop contribution
}

__global__ void count_deg(const int* __restrict__ dst, float* deg, int e) {
  int i = blockIdx.x * blockDim.x + threadIdx.x;
  if (i < e) atomicAdd(&deg[dst[i]], 1.0f);
}

__global__ void finish_dinv(float* dinv, int n) {
  int i = blockIdx.x * blockDim.x + threadIdx.x;
  if (i < n) dinv[i] = rsqrtf(fmaxf(dinv[i], 1.0f));
}

// one wave per edge; lanes stride features.  atomic scatter-add of the
// degree-normalized message into the destination row.
__global__ void edge_scatter(const int* __restrict__ src,
                             const int* __restrict__ dst,
                             const float* __restrict__ dinv,
                             const float* __restrict__ xw,
                             float* __restrict__ out, int e, int d) {
  int wave = (int)((blockIdx.x * blockDim.x + threadIdx.x) >> 5);
  int lane = (int)(threadIdx.x & 31);
  if (wave >= e) return;
  int s = src[wave];
  int t = dst[wave];
  float c = dinv[s] * dinv[t];
  const float* xs = xw + (long)s * d;
  float* od = out + (long)t * d;
  for (int f = lane; f < d; f += 32)
    atomicAdd(&od[f], xs[f] * c);
}

__global__ void bias_relu(float* __restrict__ h, const float* __restrict__ b,
                          long tot, int d) {
  long idx = (long)blockIdx.x * blockDim.x + threadIdx.x;
  if (idx < tot) h[idx] = fmaxf(h[idx] + b[idx % d], 0.0f);
}

__global__ void zero_f32(float* p, long tot) {
  long idx = (long)blockIdx.x * blockDim.x + threadIdx.x;
  if (idx < tot) p[idx] = 0.0f;
}

// segment max over sorted batch ids; h >= 0 (post-ReLU) so the int-bit
// atomicMax trick preserves float ordering.
__global__ void pool_max(const float* __restrict__ h, const int* __restrict__ batch,
                         float* __restrict__ gpool, int n, int d) {
  long idx = (long)blockIdx.x * blockDim.x + threadIdx.x;
  long tot = (long)n * d;
  if (idx < tot) {
    int i = (int)(idx / d);
    int f = (int)(idx - (long)i * d);
    int gi = batch[i];
    atomicMax((int*)&gpool[(long)gi * d + f], __float_as_int(h[idx]));
  }
}

// small dense layers: one thread per output element
__global__ void dense_rowmajor(const float* __restrict__ X, const float* __restrict__ W,
                               const float* __restrict__ b, float* __restrict__ Y,
                               int rows, int fin, int fout, int relu,
                               int ostride, int ooff) {
  int idx = blockIdx.x * blockDim.x + threadIdx.x;
  if (idx >= rows * fout) return;
  int r = idx / fout;
  int j = idx - r * fout;
  const float* x = X + (long)r * fin;
  float acc = b[j];
  for (int k = 0; k < fin; ++k) acc = fmaf(x[k], W[(long)k * fout + j], acc);
  if (relu) acc = fmaxf(acc, 0.0f);
  Y[(long)r * ostride + ooff + j] = acc;
}

// ---------------------------------------------------------------------------
extern "C" void kernel_launch(void* const* d_in, const int* in_sizes, int n_in,
                              void* d_out, int out_size, void* d_ws, size_t ws_size,
                              hipStream_t stream) {
  const float* x     = (const float*)d_in[0];
  const int*   eidx  = (const int*)d_in[1];
  const int*   batch = (const int*)d_in[2];
  const float* DDI   = (const float*)d_in[3];
  const float* W1 = (const float*)d_in[4];   const float* b1 = (const float*)d_in[5];
  const float* W2 = (const float*)d_in[6];   const float* b2 = (const float*)d_in[7];
  const float* Wg1 = (const float*)d_in[8];  const float* bg1 = (const float*)d_in[9];
  const float* Wg2 = (const float*)d_in[10]; const float* bg2 = (const float*)d_in[11];
  const float* Wf1 = (const float*)d_in[12]; const float* bf1 = (const float*)d_in[13];
  const float* Wf2 = (const float*)d_in[14]; const float* bf2 = (const float*)d_in[15];
  const float* Wf3 = (const float*)d_in[16]; const float* bf3 = (const float*)d_in[17];
  float* out = (float*)d_out;

  const int D = 64;
  const int N = in_sizes[0] / D;            // 50000
  const int E = in_sizes[1] / 2;            // 800000
  const int FDDI = 1024 + 32;               // 1056
  const int G = in_sizes[3] / FDDI;         // 64
  const int D2 = 2 * D;                     // 128
  const int HG = 1028;
  const int* src = eidx;
  const int* dst = eidx + E;

  // workspace layout (floats)
  float* ws = (float*)d_ws;
  long off = 0;
  float* dinv  = ws + off; off += N;
  float* xw1   = ws + off; off += (long)N * D;
  float* h1    = ws + off; off += (long)N * D;     // agg1 -> relu in place
  float* xw2   = ws + off; off += (long)N * D2;
  float* h2    = ws + off; off += (long)N * D2;    // agg2 -> relu in place
  float* gpool = ws + off; off += (long)G * D2;
  float* gh    = ws + off; off += (long)G * HG;
  float* dt1   = ws + off; off += (long)G * 64;
  float* dt2   = ws + off; off += (long)G * 16;
  (void)ws_size; (void)n_in; (void)out_size;

  const int T = 256;
  auto blk = [](long work, int t) { return (int)((work + t - 1) / t); };

  // ---- degree / normalization --------------------------------------------
  init_deg<<<blk(N, T), T, 0, stream>>>(dinv, N);
  count_deg<<<blk(E, T), T, 0, stream>>>(dst, dinv, E);
  finish_dinv<<<blk(N, T), T, 0, stream>>>(dinv, N);

  // ---- layer 1: xw1 = x @ W1 (fused self-loop init of h1); scatter; relu -
  {
    long waves = (long)((N + 15) / 16) * (D / 64);   // colGroups = 1
    gemm_node_wmma<64, 4><<<blk(waves * 32, T), T, 0, stream>>>(
        x, W1, xw1, h1, dinv, N, D);
  }
  edge_scatter<<<blk((long)E * 32, T), T, 0, stream>>>(src, dst, dinv, xw1, h1, E, D);
  bias_relu<<<blk((long)N * D, T), T, 0, stream>>>(h1, b1, (long)N * D, D);

  // ---- layer 2: xw2 = h1 @ W2 (fused self-loop init of h2); scatter; relu
  {
    long waves = (long)((N + 15) / 16) * (D2 / 64);  // colGroups = 2
    gemm_node_wmma<64, 4><<<blk(waves * 32, T), T, 0, stream>>>(
        h1, W2, xw2, h2, dinv, N, D2);
  }
  edge_scatter<<<blk((long)E * 32, T), T, 0, stream>>>(src, dst, dinv, xw2, h2, E, D2);
  bias_relu<<<blk((long)N * D2, T), T, 0, stream>>>(h2, b2, (long)N * D2, D2);

  // ---- global max pool ----------------------------------------------------
  zero_f32<<<blk((long)G * D2, T), T, 0, stream>>>(gpool, (long)G * D2);
  pool_max<<<blk((long)N * D2, T), T, 0, stream>>>(h2, batch, gpool, N, D2);

  // ---- graph head: gh = relu(gpool @ Wg1 + bg1); ppi = gh @ Wg2 + bg2 ----
  {
    long waves = (long)((G + 15) / 16) * ((HG + 15) / 16);
    gemm_wmma_f32<<<blk(waves * 32, T), T, 0, stream>>>(gpool, Wg1, gh, G, HG, D2);
  }
  bias_relu<<<blk((long)G * HG, T), T, 0, stream>>>(gh, bg1, (long)G * HG, HG);
  dense_rowmajor<<<blk((long)G * 32, T), T, 0, stream>>>(gh, Wg2, bg2, out,
                                                         G, HG, 32, 0, 33, 0);

  // ---- DDI head: relu chains, final column 32 of output ------------------
  dense_rowmajor<<<blk((long)G * 64, T), T, 0, stream>>>(DDI, Wf1, bf1, dt1,
                                                         G, FDDI, 64, 1, 64, 0);
  dense_rowmajor<<<blk((long)G * 16, T), T, 0, stream>>>(dt1, Wf2, bf2, dt2,
                                                         G, 64, 16, 1, 16, 0);
  dense_rowmajor<<<blk((long)G, T), T, 0, stream>>>(dt2, Wf3, bf3, out,
                                                    G, 16, 1, 1, 33, 32);
}